// SSD_70300024701369
// MI455X (gfx1250) — compile-verified
//
#include <hip/hip_runtime.h>
#include <hip/hip_bf16.h>

// PointPillars PFNv2 for MI455X (gfx1250, wave32).
// Memory-bound: 256 MB output scatter dominates (23.3 TB/s HBM floor ~11us).
// Tiny [480K x 10] @ [10 x 64] GEMM runs on V_WMMA_F32_16X16X4_F32 (fp32,
// exact vs reference), 16 points per wave, BN scale folded into a per-block
// LDS copy of W. Segment-max = integer atomicMax on float bits (values >= 0).

#define GRID_ 500
#define NSEG_ 250000          // GRID_*GRID_
#define NPTS_ 120000          // points per batch (divisible by 16)
#define BATCH_ 4
#define RES_ 0.16f
#define EPS_ 1e-5f
#define MAXPTS_ 100.0f
#define COUT_ 64

typedef __attribute__((ext_vector_type(2))) float v2f;
typedef __attribute__((ext_vector_type(8))) float v8f;

// ---------------------------------------------------------------------------
// Pass 1: per-point pillar index + pillar accumulators (cnt, sx, sy, sz).
// ---------------------------------------------------------------------------
__global__ void pf_accum(const float* __restrict__ pts,
                         float* __restrict__ cnt, float* __restrict__ sx,
                         float* __restrict__ sy, float* __restrict__ sz,
                         int* __restrict__ flatArr) {
  int i = blockIdx.x * blockDim.x + threadIdx.x;
  if (i >= BATCH_ * NPTS_) return;
  int b = i / NPTS_;
  const float* p = pts + (size_t)i * 5;
  float x = p[0], y = p[1], z = p[2];
  int xi = (int)floorf((x + 40.0f) / RES_);
  xi = min(max(xi, 0), GRID_ - 1);
  int zi = (int)floorf((80.0f - z) / RES_);
  zi = min(max(zi, 0), GRID_ - 1);
  int flat = zi * GRID_ + xi;
  flatArr[i] = flat;
  int o = b * NSEG_ + flat;
  atomicAdd(&cnt[o], 1.0f);
  atomicAdd(&sx[o], x);
  atomicAdd(&sy[o], y);
  atomicAdd(&sz[o], z);
}

// ---------------------------------------------------------------------------
// Pass 2: feature build + WMMA GEMM + ReLU + segment-max scatter.
// One wave32 per 16-point tile. D(16x64) = A(16x12) x B(12x64) via 3 K-steps
// of V_WMMA_F32_16X16X4_F32 over 4 N-tiles.
//
// ISA 7.12.2 layouts (32-bit, wave32):
//   A 16x4 : lane&15 = M; lane<16 holds K {k0,k0+1}, lane>=16 holds {k0+2,k0+3}
//   B 4x16 : mirrored; lane&15 = N; same K split by lane half
//   D 16x16: lane&15 = N; VGPR v -> M = v + 8*(lane>=16)
// ---------------------------------------------------------------------------
__global__ void pf_wmma(const float* __restrict__ pts,
                        const float* __restrict__ W,
                        const float* __restrict__ gamma,
                        const float* __restrict__ beta,
                        const float* __restrict__ rmean,
                        const float* __restrict__ rvar,
                        const float* __restrict__ cnt,
                        const float* __restrict__ sx,
                        const float* __restrict__ sy,
                        const float* __restrict__ sz,
                        const int* __restrict__ flatArr,
                        float* __restrict__ out) {
  // Folded B-matrix [n][k] (scale[n]*W[n][k], k padded 10->12) + bias, in LDS.
  __shared__ float Bs[COUT_][12];
  __shared__ float Sh[COUT_];
  int tid = threadIdx.x;
  if (tid < COUT_) {
    float sc = gamma[tid] * rsqrtf(rvar[tid] + EPS_);
    Sh[tid] = beta[tid] - rmean[tid] * sc;
#pragma unroll
    for (int k = 0; k < 10; ++k) Bs[tid][k] = W[tid * 10 + k] * sc;
    Bs[tid][10] = 0.0f;
    Bs[tid][11] = 0.0f;
  }
  __syncthreads();

  const int tilesPerBatch = NPTS_ / 16;
  int wave = blockIdx.x * (blockDim.x >> 5) + (threadIdx.x >> 5);
  int lane = threadIdx.x & 31;
  int b = wave / tilesPerBatch;
  int tbase = (wave % tilesPerBatch) * 16;
  bool hi = lane >= 16;      // upper lane half: K+2 for A/B, M+8 for D
  int nlo = lane & 15;

  // --- Build the 10 features for point M = lane&15 -------------------------
  size_t pIdx = (size_t)b * NPTS_ + tbase + nlo;
  const float* p = pts + pIdx * 5;
  float x = p[0], y = p[1], z = p[2], r = p[3], c = p[4];
  int flat = flatArr[pIdx];
  int o = b * NSEG_ + flat;
  float cn = fmaxf(cnt[o], 1.0f);
  float xm = sx[o] / cn, ym = sy[o] / cn, zm = sz[o] / cn;
  int xi = flat % GRID_;
  int zi = flat / GRID_;
  float xc = -40.0f + 0.5f * RES_ + (float)xi * RES_;
  float zc =  80.0f - 0.5f * RES_ - (float)zi * RES_;
  float f0 = x, f1 = y, f2 = z, f3 = r, f4 = c;
  float f5 = x - xm, f6 = y - ym, f7 = z - zm, f8 = x - xc, f9 = z - zc;

  // A-matrix fragments (nt-invariant): literal-index selects only.
  v2f a0, a4, a8;
  a0.x = hi ? f2 : f0;    a0.y = hi ? f3 : f1;
  a4.x = hi ? f6 : f4;    a4.y = hi ? f7 : f5;
  a8.x = hi ? 0.0f : f8;  a8.y = hi ? 0.0f : f9;   // K=10,11 are zero pad

  // Pillar indices for the 8 D-rows this lane writes (two b128 loads).
  int mrow = hi ? 8 : 0;
  const int4* fp = (const int4*)(flatArr + (size_t)b * NPTS_ + tbase + mrow);
  int4 fA = fp[0];
  int4 fB = fp[1];
  int fmv[8] = {fA.x, fA.y, fA.z, fA.w, fB.x, fB.y, fB.z, fB.w};

  int khalf = hi ? 2 : 0;

#pragma unroll
  for (int nt = 0; nt < 4; ++nt) {
    int n = nt * 16 + nlo;
    // B fragments from LDS: pairs (k,k+1) contiguous, 8B-aligned (k even).
    const float* brow = &Bs[n][khalf];
    v2f b0 = *(const v2f*)(brow);
    v2f b4 = *(const v2f*)(brow + 4);
    v2f b8 = *(const v2f*)(brow + 8);
    float sh = Sh[n];

    v8f acc = {};
    acc = __builtin_amdgcn_wmma_f32_16x16x4_f32(false, a0, false, b0,
                                                (short)0, acc, false, false);
    acc = __builtin_amdgcn_wmma_f32_16x16x4_f32(false, a4, false, b4,
                                                (short)0, acc, false, false);
    acc = __builtin_amdgcn_wmma_f32_16x16x4_f32(false, a8, false, b8,
                                                (short)0, acc, false, false);

    // ReLU + segment-max scatter (values >= 0 => int order == float order).
#pragma unroll
    for (int v = 0; v < 8; ++v) {
      float val = fmaxf(acc[v] + sh, 0.0f);
      int* dst = (int*)(out + ((size_t)b * COUT_ + n) * NSEG_ + fmv[v]);
      atomicMax(dst, __float_as_int(val));
    }
  }
}

// ---------------------------------------------------------------------------
// Pass 3: apply pad_val where 0 < cnt < MAXPTS (other cases already correct:
// empty pillars stayed 0; cnt>=MAXPTS keeps raw segmax).
// ---------------------------------------------------------------------------
__global__ void pf_finalize(const float* __restrict__ cnt,
                            const float* __restrict__ gamma,
                            const float* __restrict__ beta,
                            const float* __restrict__ rmean,
                            const float* __restrict__ rvar,
                            float* __restrict__ out) {
  size_t i = (size_t)blockIdx.x * blockDim.x + threadIdx.x;
  const size_t TOT = (size_t)BATCH_ * COUT_ * NSEG_;
  if (i >= TOT) return;
  int pp = (int)(i % NSEG_);
  int cc = (int)((i / NSEG_) % COUT_);
  int bb = (int)(i / ((size_t)NSEG_ * COUT_));
  float cn = cnt[bb * NSEG_ + pp];
  if (cn > 0.0f && cn < MAXPTS_) {
    float sc = gamma[cc] * rsqrtf(rvar[cc] + EPS_);
    float pv = fmaxf(beta[cc] - rmean[cc] * sc, 0.0f);
    float cur = out[i];
    if (pv > cur) out[i] = pv;
  }
}

// ---------------------------------------------------------------------------
extern "C" void kernel_launch(void* const* d_in, const int* in_sizes, int n_in,
                              void* d_out, int out_size, void* d_ws,
                              size_t ws_size, hipStream_t stream) {
  const float* pts   = (const float*)d_in[0];  // [4,120000,5]
  const float* W     = (const float*)d_in[1];  // [64,10]
  const float* gamma = (const float*)d_in[2];  // [64]
  const float* beta  = (const float*)d_in[3];  // [64]
  const float* rmean = (const float*)d_in[4];  // [64]
  const float* rvar  = (const float*)d_in[5];  // [64]
  float* out = (float*)d_out;                  // [4,64,500,500]

  // Workspace layout: cnt | sx | sy | sz (each B*NSEG f32) | flat (B*N i32)
  float* wsf = (float*)d_ws;
  const size_t seg = (size_t)BATCH_ * NSEG_;
  float* cnt = wsf;
  float* sx  = wsf + seg;
  float* sy  = wsf + 2 * seg;
  float* sz  = wsf + 3 * seg;
  int* flat  = (int*)(wsf + 4 * seg);

  hipMemsetAsync(d_ws, 0, 4 * seg * sizeof(float), stream);
  hipMemsetAsync(d_out, 0, (size_t)out_size * sizeof(float), stream);

  int np = BATCH_ * NPTS_;
  pf_accum<<<(np + 255) / 256, 256, 0, stream>>>(pts, cnt, sx, sy, sz, flat);

  int ntiles = BATCH_ * (NPTS_ / 16);          // 30000 waves, 8 waves/block
  pf_wmma<<<(ntiles + 7) / 8, 256, 0, stream>>>(pts, W, gamma, beta, rmean,
                                                rvar, cnt, sx, sy, sz, flat,
                                                out);

  size_t tot = (size_t)BATCH_ * COUT_ * NSEG_;
  pf_finalize<<<(int)((tot + 255) / 256), 256, 0, stream>>>(cnt, gamma, beta,
                                                            rmean, rvar, out);
}